// MixModel_20486994002408
// MI455X (gfx1250) — compile-verified
//
#include <hip/hip_runtime.h>

// CDNA5 / gfx1250, wave32. Matrix op: V_WMMA_F32_16X16X4_F32 (D = A(16x4) * B(4x16) + C).
// Layouts (ISA 7.12.2):
//   A: lane = M (0-15 both halves); VGPR0 = K0 | K2 (lane halves), VGPR1 = K1 | K3
//   B: lane = N column;             VGPR0 = K0 | K2,               VGPR1 = K1 | K3
//   C/D: VGPR r -> rows r (lanes 0-15) and r+8 (lanes 16-31), lane = N column.

typedef __attribute__((ext_vector_type(2))) float v2f;
typedef __attribute__((ext_vector_type(8))) float v8f;

#define WPB 4  // waves per 128-thread block

__device__ __forceinline__ float swapx16(float x) {
  // ds_swizzle group-of-32, xor=0x10, and=0x1f  -> swap lanes L <-> L^16
  return __int_as_float(__builtin_amdgcn_ds_swizzle(__float_as_int(x), 0x401F));
}

__device__ __forceinline__ v8f wmma4(v2f a, v2f b, v8f c) {
  return __builtin_amdgcn_wmma_f32_16x16x4_f32(false, a, false, b, (short)0, c, false, false);
}

__global__ void __launch_bounds__(128)
mixmodel_kernel(const float* __restrict__ u, const float* __restrict__ reg,
                const float* __restrict__ n1_w0, const float* __restrict__ n1_b0,
                const float* __restrict__ n1_w1, const float* __restrict__ n1_b1,
                const float* __restrict__ n1_w2, const float* __restrict__ n1_b2,
                const float* __restrict__ n2_w0, const float* __restrict__ n2_b0,
                const float* __restrict__ n2_w1, const float* __restrict__ n2_b1,
                const float* __restrict__ n2_w2, const float* __restrict__ n2_b2,
                const float* __restrict__ n3_w0, const float* __restrict__ n3_b0,
                const float* __restrict__ n3_w1, const float* __restrict__ n3_b1,
                const float* __restrict__ n3_w2, const float* __restrict__ n3_b2,
                float* __restrict__ out, int N)
{
  __shared__ float ldsW[11 * 64];     // 11 zero-padded 16x4 A-matrix images
  __shared__ float ldsIn[WPB * 576];  // per-wave [sample 0..15][feature 0..35]
  __shared__ float ldsOut[WPB * 576];

  const int tid  = threadIdx.x;
  const int lane = tid & 31;
  const int wid  = tid >> 5;
  const int sBase = (blockIdx.x * WPB + wid) * 16;

  // ---- build A-matrix images in LDS (thread 0 of block; tiny, constant work) ----
  if (tid == 0) {
    for (int x = 0; x < 11 * 64; ++x) ldsW[x] = 0.f;
    // m0: layer0 for mlp1 (rows 0-2, K offset 0) + mlp2 (rows 3-5, K offset 1)
    for (int r = 0; r < 3; ++r)
      for (int k = 0; k < 3; ++k) {
        ldsW[0 * 64 + r * 4 + k]           = n1_w0[r * 3 + k];
        ldsW[0 * 64 + (3 + r) * 4 + k + 1] = n2_w0[r * 3 + k];
      }
    // m1: layer0 mlp3 (4x4)
    for (int r = 0; r < 4; ++r)
      for (int k = 0; k < 4; ++k) ldsW[1 * 64 + r * 4 + k] = n3_w0[r * 4 + k];
    // m2/m3: layer1 mlp1/mlp2 (6x3); m4: layer1 mlp3 (6x4)
    for (int r = 0; r < 6; ++r) {
      for (int k = 0; k < 3; ++k) {
        ldsW[2 * 64 + r * 4 + k] = n1_w1[r * 3 + k];
        ldsW[3 * 64 + r * 4 + k] = n2_w1[r * 3 + k];
      }
      for (int k = 0; k < 4; ++k) ldsW[4 * 64 + r * 4 + k] = n3_w1[r * 4 + k];
    }
    // m5/m6: layer2 mlp1 (3x6 split K 0-3 / 4-5); m7/m8: mlp2; m9/m10: mlp3 (2x6)
    for (int r = 0; r < 3; ++r) {
      for (int k = 0; k < 4; ++k) { ldsW[5 * 64 + r * 4 + k] = n1_w2[r * 6 + k];
                                    ldsW[7 * 64 + r * 4 + k] = n2_w2[r * 6 + k]; }
      for (int k = 0; k < 2; ++k) { ldsW[6 * 64 + r * 4 + k] = n1_w2[r * 6 + 4 + k];
                                    ldsW[8 * 64 + r * 4 + k] = n2_w2[r * 6 + 4 + k]; }
    }
    for (int r = 0; r < 2; ++r) {
      for (int k = 0; k < 4; ++k) ldsW[9 * 64 + r * 4 + k]  = n3_w2[r * 6 + k];
      for (int k = 0; k < 2; ++k) ldsW[10 * 64 + r * 4 + k] = n3_w2[r * 6 + 4 + k];
    }
  }

  // ---- stage this wave's 16 samples into LDS (coalesced; clamp for tail) ----
  float* inS = ldsIn + wid * 576;
  if (sBase < N) {
    for (int e = lane; e < 576; e += 32) {
      int s = sBase + e / 36;
      if (s >= N) s = N - 1;
      inS[e] = u[(size_t)s * 36 + (e - (e / 36) * 36)];
    }
  }
  __syncthreads();
  if (sBase >= N) return;  // whole-wave uniform; no barriers below

  const int  M    = lane & 15;             // sample column / A row
  const int  kh   = (lane & 16) ? 2 : 0;   // K half for A/B fragments
  const bool isLo = (lane & 16) == 0;

  auto ldA = [&](int m) -> v2f {
    v2f a; a.x = ldsW[m * 64 + M * 4 + kh]; a.y = ldsW[m * 64 + M * 4 + kh + 1]; return a;
  };
  const v2f A012 = ldA(0), A03 = ldA(1);
  const v2f A11 = ldA(2), A12w = ldA(3), A13 = ldA(4);
  const v2f A21lo = ldA(5), A21hi = ldA(6);
  const v2f A22lo = ldA(7), A22hi = ldA(8);
  const v2f A23lo = ldA(9), A23hi = ldA(10);

  const int sline = M * 36;
  float* outS = ldsOut + wid * 576;
  const float reg0 = reg[0];
  const v8f z = {0.f, 0.f, 0.f, 0.f, 0.f, 0.f, 0.f, 0.f};

  // u1[q] -> u index: q even -> 3*(q/2); q odd -> 3*(q/2)+1 (wrap mod 24)
  auto u1feat = [&](int q) -> float {
    if (q < 0) q += 24;
    if (q >= 24) q -= 24;
    int ui = (q >> 1) * 3 + (q & 1);
    return inS[sline + ui];
  };

#pragma unroll
  for (int i = 0; i < 12; ++i) {
    const int j = 2 * i;
    // layer-0 B fragments. Shared middle fragment: W1.B1 == W2.B0.
    const float fa = u1feat(j - 1 + kh);  // K0|K2 of window [j-1..j+2]
    const float fb = u1feat(j + kh);      // K1|K3 of window 1 == K0|K2 of window 2
    const float fc = u1feat(j + 1 + kh);  // K1|K3 of window [j..j+3]
    v2f B1; B1.x = fa; B1.y = fb;
    v2f B2; B2.x = fb; B2.y = fc;

    v8f d12 = wmma4(A012, B1, z);   // rows 0-2: mlp1 L0, rows 3-5: mlp2 L0
    v8f d3  = wmma4(A03,  B2, z);   // rows 0-3: mlp3 L0

    float h1[3], h2[3], h3[4];
#pragma unroll
    for (int k = 0; k < 3; ++k) {
      h1[k] = fmaxf(d12[k]     + n1_b0[k], 0.f);
      h2[k] = fmaxf(d12[3 + k] + n2_b0[k], 0.f);
    }
#pragma unroll
    for (int k = 0; k < 4; ++k) h3[k] = fmaxf(d3[k] + n3_b0[k], 0.f);

    // layer-1 B: K2/K3 rows come from lanes 0-15 via SWAPX16 (K=3 pads A col3 = 0)
    const float t1 = swapx16(h1[2]);
    const float t2 = swapx16(h2[2]);
    const float t3a = swapx16(h3[2]);
    const float t3b = swapx16(h3[3]);
    v2f b1; b1.x = isLo ? h1[0] : t1;  b1.y = h1[1];
    v2f b2; b2.x = isLo ? h2[0] : t2;  b2.y = h2[1];
    v2f b3; b3.x = isLo ? h3[0] : t3a; b3.y = isLo ? h3[1] : t3b;

    v8f e1 = wmma4(A11,  b1, z);
    v8f e2 = wmma4(A12w, b2, z);
    v8f e3 = wmma4(A13,  b3, z);

    float g1[6], g2[6], g3[6];
#pragma unroll
    for (int k = 0; k < 6; ++k) {
      g1[k] = fmaxf(e1[k] + n1_b1[k], 0.f);
      g2[k] = fmaxf(e2[k] + n2_b1[k], 0.f);
      g3[k] = fmaxf(e3[k] + n3_b1[k], 0.f);
    }

    // layer-2: K=6 -> chain two WMMAs (K 0-3, then K 4-5 with C-accumulate)
    const float s1a = swapx16(g1[2]), s1b = swapx16(g1[3]);
    const float s2a = swapx16(g2[2]), s2b = swapx16(g2[3]);
    const float s3a = swapx16(g3[2]), s3b = swapx16(g3[3]);
    v2f c1lo; c1lo.x = isLo ? g1[0] : s1a; c1lo.y = isLo ? g1[1] : s1b;
    v2f c2lo; c2lo.x = isLo ? g2[0] : s2a; c2lo.y = isLo ? g2[1] : s2b;
    v2f c3lo; c3lo.x = isLo ? g3[0] : s3a; c3lo.y = isLo ? g3[1] : s3b;
    v2f c1hi; c1hi.x = g1[4]; c1hi.y = g1[5];  // A hi cols K2/K3 are zero
    v2f c2hi; c2hi.x = g2[4]; c2hi.y = g2[5];
    v2f c3hi; c3hi.x = g3[4]; c3hi.y = g3[5];

    v8f f1 = wmma4(A21lo, c1lo, z); f1 = wmma4(A21hi, c1hi, f1);
    v8f f2 = wmma4(A22lo, c2lo, z); f2 = wmma4(A22hi, c2hi, f2);
    v8f f3 = wmma4(A23lo, c3lo, z); f3 = wmma4(A23hi, c3hi, f3);

    // final combine: lanes 0-15 hold all neuron rows for their own sample
    if (isLo) {
      const float uA  = inS[sline + ((3 * i + 35) % 36)];
      const float uB  = inS[sline + 3 * i + 2];
      const float u0  = inS[sline + 3 * i];
      const float u1v = inS[sline + 3 * i + 1];
      const float o10 = f1[0] + n1_b2[0], o11 = f1[1] + n1_b2[1], o12 = f1[2] + n1_b2[2];
      const float o20 = f2[0] + n2_b2[0], o21 = f2[1] + n2_b2[1], o22 = f2[2] + n2_b2[2];
      const float o30 = f3[0] + n3_b2[0], o31 = f3[1] + n3_b2[1];
      const float r0 = fmaf(u0,  reg[1 + 3 * i], reg0);
      const float r1 = fmaf(u1v, reg[2 + 3 * i], reg0);
      const float r2 = fmaf(uB,  reg[3 + 3 * i], reg0);
      outS[sline + 3 * i]     = r0 + o10 + o11 * uA + o12 * uB;
      outS[sline + 3 * i + 1] = r1 + o20 + o21 * uA + o22 * uB;
      outS[sline + 3 * i + 2] = r2 + o30 + o31 * uB;
    }
  }

  // ---- coalesced writeback (per wave; same-wave LDS ordering guarantees) ----
  for (int e = lane; e < 576; e += 32) {
    int s = sBase + e / 36;
    if (s < N) out[(size_t)sBase * 36 + e] = outS[e];
  }
}

extern "C" void kernel_launch(void* const* d_in, const int* in_sizes, int n_in,
                              void* d_out, int out_size, void* d_ws, size_t ws_size,
                              hipStream_t stream) {
  (void)n_in; (void)out_size; (void)d_ws; (void)ws_size;
  const float* u   = (const float*)d_in[1];
  const float* reg = (const float*)d_in[2];
  const int N = in_sizes[1] / 36;
  const int nWaves  = (N + 15) / 16;
  const int nBlocks = (nWaves + WPB - 1) / WPB;
  mixmodel_kernel<<<nBlocks, 128, 0, stream>>>(
      u, reg,
      (const float*)d_in[3],  (const float*)d_in[4],  (const float*)d_in[5],
      (const float*)d_in[6],  (const float*)d_in[7],  (const float*)d_in[8],
      (const float*)d_in[9],  (const float*)d_in[10], (const float*)d_in[11],
      (const float*)d_in[12], (const float*)d_in[13], (const float*)d_in[14],
      (const float*)d_in[15], (const float*)d_in[16], (const float*)d_in[17],
      (const float*)d_in[18], (const float*)d_in[19], (const float*)d_in[20],
      (float*)d_out, N);
}